// Loss_16681652978188
// MI455X (gfx1250) — compile-verified
//
#include <hip/hip_runtime.h>
#include <math.h>

// ---------------- problem constants ----------------
#define SGRID        7
#define DCELL        30                    // B*5 + C = 2*5 + 20
#define BATCH_N      16384
#define CELLS_TOTAL  (16384 * 49)          // 802816
#define CELLS_PER_BLK 256
#define NBLOCKS      (CELLS_TOTAL / CELLS_PER_BLK)   // 3136, exact (no tail)
#define BLK_FLOATS   (CELLS_PER_BLK * DCELL)         // 7680 floats = 30 KB
#define L_COORD_C    5.0f
#define L_NOOBJ_C    0.5f
#define EPS_C        1e-6f

typedef float v2f __attribute__((ext_vector_type(2)));
typedef float v8f __attribute__((ext_vector_type(8)));

// ---- CDNA5 async global->LDS staging (ASYNCcnt path), with safe fallback ----
#if defined(__has_builtin)
#  if __has_builtin(__builtin_amdgcn_global_load_async_to_lds_b128) && \
      __has_builtin(__builtin_amdgcn_s_wait_asynccnt)
#    define USE_ASYNC_LDS 1
#  endif
#endif
#ifndef USE_ASYNC_LDS
#  define USE_ASYNC_LDS 0
#endif

#if USE_ASYNC_LDS
// Builtin expects: (int __attribute__((vector_size(16))) __device__ *gsrc,  <lds dst>, imm, imm)
typedef int v4i_vs __attribute__((vector_size(16)));
typedef __attribute__((address_space(1))) v4i_vs glob_v4i;   // prints as "__device__"
typedef __attribute__((address_space(3))) v4i_vs lds_v4i;    // prints as "__shared__"
#endif

// ---------------- per-cell YOLO loss (branchless, mirrors reference) ----------------
__device__ __forceinline__ float cell_loss(const float* __restrict__ p,
                                           const float* __restrict__ t) {
    const float invS = 1.0f / (float)SGRID;
    const float t4  = t[4];
    const float obj = (t4 > 0.0f)  ? 1.0f : 0.0f;
    const float noo = (t4 == 0.0f) ? 1.0f : 0.0f;

    // no-object confidence loss (both conf slots 4 and 9)
    const float d4 = p[4] - t4;
    const float d9 = p[9] - t[9];
    const float noobj_loss = d4 * d4 + d9 * d9;

    // class loss (elements 10..29)
    float cls = 0.0f;
#pragma unroll
    for (int i = 10; i < DCELL; ++i) {
        const float d = p[i] - t[i];
        cls += d * d;
    }

    // boxes -> xyxy corners + areas (areas from corner diffs, as reference does)
    float px1[2], py1[2], px2[2], py2[2], pa[2];
    float tx1[2], ty1[2], tx2[2], ty2[2], ta[2];
#pragma unroll
    for (int b = 0; b < 2; ++b) {
        const float* bp = p + 5 * b;
        const float* bt = t + 5 * b;
        float cx = bp[0] * invS, cy = bp[1] * invS, w = bp[2], h = bp[3];
        px1[b] = cx - 0.5f * w;  py1[b] = cy - 0.5f * h;
        px2[b] = cx + 0.5f * w;  py2[b] = cy + 0.5f * h;
        pa[b]  = (px2[b] - px1[b]) * (py2[b] - py1[b]);
        cx = bt[0] * invS; cy = bt[1] * invS; w = bt[2]; h = bt[3];
        tx1[b] = cx - 0.5f * w;  ty1[b] = cy - 0.5f * h;
        tx2[b] = cx + 0.5f * w;  ty2[b] = cy + 0.5f * h;
        ta[b]  = (tx2[b] - tx1[b]) * (ty2[b] - ty1[b]);
    }

    // pairwise IoU: iou[i][j] = pred box i vs target box j
    float iou[2][2];
#pragma unroll
    for (int i = 0; i < 2; ++i) {
#pragma unroll
        for (int j = 0; j < 2; ++j) {
            const float lx = fmaxf(px1[i], tx1[j]);
            const float ly = fmaxf(py1[i], ty1[j]);
            const float rx = fminf(px2[i], tx2[j]);
            const float ry = fminf(py2[i], ty2[j]);
            const float w  = fmaxf(rx - lx, 0.0f);
            const float h  = fmaxf(ry - ly, 0.0f);
            const float inter = w * h;
            iou[i][j] = inter / (pa[i] + ta[j] - inter);
        }
    }

    // argmax over pred-box axis, first-occurrence tie break (strict > picks 1)
    const int idx0 = (iou[1][0] > iou[0][0]) ? 1 : 0;
    const int idx1 = (iou[1][1] > iou[0][1]) ? 1 : 0;
    const float mx0 = idx0 ? iou[1][0] : iou[0][0];
    const float mx1 = idx1 ? iou[1][1] : iou[0][1];

    float contain = 0.0f, loc = 0.0f;
#pragma unroll
    for (int b = 0; b < 2; ++b) {
        const float r     = ((idx0 == b) || (idx1 == b)) ? 1.0f : 0.0f;
        const float resp  = r * obj;
        // reference: conf_t set from idx0 match, then overridden by idx1 match
        const float conft = (idx1 == b) ? mx1 : ((idx0 == b) ? mx0 : 0.0f);
        const float dc = p[5 * b + 4] - conft;
        contain += resp * dc * dc;
        const float dx = resp * (p[5 * b + 0] - t[5 * b + 0]);
        const float dy = resp * (p[5 * b + 1] - t[5 * b + 1]);
        loc += dx * dx + dy * dy;
        const float sw = sqrtf(resp * p[5 * b + 2] + EPS_C) - sqrtf(resp * t[5 * b + 2] + EPS_C);
        const float sh = sqrtf(resp * p[5 * b + 3] + EPS_C) - sqrtf(resp * t[5 * b + 3] + EPS_C);
        loc += sw * sw + sh * sh;
    }

    return L_COORD_C * loc + contain + L_NOOBJ_C * (noo * noobj_loss) + obj * cls;
}

// ---- single-instruction lane exchange: ds_swizzle group-of-32, XOR pattern ----
template <int XMASK>
__device__ __forceinline__ float swz_xor_add(float x) {
    const int pat = (XMASK << 10) | 0x1f;   // offset[14:10]=xor, [9:5]=or=0, [4:0]=and=0x1f
    int o = __builtin_amdgcn_ds_swizzle(__float_as_int(x), pat);
    return x + __int_as_float(o);
}

// ---------------- wave32 reduction: one V_WMMA_F32_16X16X4_F32 (32 -> 16) + 4 swizzles ----------------
// A = ones(16x4), B VGPR0 = lane partials => rows K=0 (lanes 0-15) / K=1 (lanes 16-31),
// B VGPR1 = 0.  D[m][n] = partial[n] + partial[n+16] replicated in every lane (n = lane & 15).
// Requires EXEC all-ones: called only from uniform full-block control flow.
__device__ __forceinline__ float wave_reduce_wmma(float v) {
    v2f a; a[0] = 1.0f; a[1] = 1.0f;
    v2f b; b[0] = v;    b[1] = 0.0f;
    v8f c = {};
    v8f d = __builtin_amdgcn_wmma_f32_16x16x4_f32(
        /*neg_a=*/false, a, /*neg_b=*/false, b,
        /*c_mod=*/(short)0, c, /*reuse_a=*/false, /*reuse_b=*/false);
    float s = d[0];
    s = swz_xor_add<8>(s);
    s = swz_xor_add<4>(s);
    s = swz_xor_add<2>(s);
    s = swz_xor_add<1>(s);
    return s;   // full wave sum in every lane
}

// ---------------- stage 1: per-block tile -> LDS -> per-cell loss -> block partial ----------------
__global__ __launch_bounds__(256) void yolo_loss_stage1(const float* __restrict__ pred,
                                                        const float* __restrict__ tgt,
                                                        float* __restrict__ partials) {
    __shared__ float sp[BLK_FLOATS];   // 30 KB
    __shared__ float st[BLK_FLOATS];   // 30 KB
    __shared__ float wpart[8];

    const int    tid  = threadIdx.x;
    const size_t base = (size_t)blockIdx.x * (size_t)BLK_FLOATS;
    const float4* gp = (const float4*)(pred + base);   // 30720-byte aligned tile
    const float4* gt = (const float4*)(tgt  + base);

    const int NV4 = BLK_FLOATS / 4;    // 1920 float4 per tensor tile

#if USE_ASYNC_LDS
    // CDNA5 path: DMA straight into LDS (global_load_async_to_lds_b128, ASYNCcnt)
    for (int i = tid; i < NV4; i += 256) {
        __builtin_amdgcn_global_load_async_to_lds_b128(
            (glob_v4i*)(gp + i), (lds_v4i*)(sp + 4 * i), 0, 0);
        __builtin_amdgcn_global_load_async_to_lds_b128(
            (glob_v4i*)(gt + i), (lds_v4i*)(st + 4 * i), 0, 0);
    }
    __builtin_amdgcn_s_wait_asynccnt(0);   // this wave's async copies are in LDS
#else
    __builtin_prefetch(gp, 0, 1);          // global_prefetch_b8
    __builtin_prefetch(gt, 0, 1);
    float4* lsp = (float4*)sp;
    float4* lst = (float4*)st;
    for (int i = tid; i < NV4; i += 256) {
        lsp[i] = gp[i];                    // global_load_b128 -> ds_store_b128
        lst[i] = gt[i];
    }
#endif
    __syncthreads();

    const float loss = cell_loss(sp + tid * DCELL, st + tid * DCELL);

    const float wsum = wave_reduce_wmma(loss);
    if ((tid & 31) == 0) wpart[tid >> 5] = wsum;
    __syncthreads();
    if (tid == 0) {
        float s = 0.0f;
#pragma unroll
        for (int i = 0; i < 8; ++i) s += wpart[i];
        partials[blockIdx.x] = s;
    }
}

// ---------------- stage 2: fold block partials, scale by 1/N ----------------
__global__ __launch_bounds__(256) void yolo_loss_stage2(const float* __restrict__ partials,
                                                        int n,
                                                        float* __restrict__ out) {
    __shared__ float wpart[8];
    float acc = 0.0f;
    for (int i = threadIdx.x; i < n; i += 256) acc += partials[i];

    const float wsum = wave_reduce_wmma(acc);
    if ((threadIdx.x & 31) == 0) wpart[threadIdx.x >> 5] = wsum;
    __syncthreads();
    if (threadIdx.x == 0) {
        float s = 0.0f;
#pragma unroll
        for (int i = 0; i < 8; ++i) s += wpart[i];
        out[0] = s * (1.0f / (float)BATCH_N);
    }
}

// ---------------- launcher ----------------
extern "C" void kernel_launch(void* const* d_in, const int* in_sizes, int n_in,
                              void* d_out, int out_size, void* d_ws, size_t ws_size,
                              hipStream_t stream) {
    const float* pred = (const float*)d_in[0];   // (16384,7,7,30) f32
    const float* tgt  = (const float*)d_in[1];   // (16384,7,7,30) f32
    float* partials   = (float*)d_ws;            // NBLOCKS floats (12.25 KB) of scratch
    float* out        = (float*)d_out;           // scalar f32

    yolo_loss_stage1<<<NBLOCKS, 256, 0, stream>>>(pred, tgt, partials);
    yolo_loss_stage2<<<1, 256, 0, stream>>>(partials, NBLOCKS, out);
}